// MultiEmbed_30520037606027
// MI455X (gfx1250) — compile-verified
//
#include <hip/hip_runtime.h>

typedef float v2f __attribute__((ext_vector_type(2)));
typedef float v4f __attribute__((ext_vector_type(4)));
typedef float v8f __attribute__((ext_vector_type(8)));

namespace {

constexpr int kB = 32;
constexpr int kN = 128;
constexpr int kE = 64;
constexpr int kHours = 168;

// -------- joint = emb_t[tix] + emb_l[l] + emb_u[u] ------------------------
// 65536 threads, one float4 of E per thread. Pure gather + add, 1 MB output.
__global__ __launch_bounds__(256) void joint_kernel(
    const int* __restrict__ traj,
    const float* __restrict__ emb_t,
    const float* __restrict__ emb_l,
    const float* __restrict__ emb_u,
    float* __restrict__ out) {
  const int tid = blockIdx.x * 256 + threadIdx.x;
  const int r  = tid >> 4;            // (b*128 + n), 0..4095
  const int e0 = (tid & 15) << 2;     // 0..60 step 4
  const int u = traj[r * 3 + 0];
  const int l = traj[r * 3 + 1];
  const int t = traj[r * 3 + 2];
  const int tix = (t + kHours - 1) % kHours + 1;   // Python (t-1)%168 + 1
  const v4f vt = *(const v4f*)(emb_t + (size_t)tix * kE + e0);
  const v4f vl = *(const v4f*)(emb_l + (size_t)l   * kE + e0);
  const v4f vu = *(const v4f*)(emb_u + (size_t)u   * kE + e0);
  *(v4f*)(out + (size_t)r * kE + e0) = vt + vl + vu;
}

// -------- delta: rank-8 update per 16x64 tile via V_WMMA_F32_16X16X4_F32 --
// delta[b,i,j,e] = (esl*vsu + esu*vsl)/100 + (etl*vtu + etu*vtl)/1000
// with e** selected by mask m. Formulated as A(16x8) x B(8x64):
//   A row j = [(1-m)*vsu,(1-m)*vsl,(1-m)*vtu,(1-m)*vtl, m*vsu,m*vsl,m*vtu,m*vtl]
//   B rows  = [sl0,su0,tl0,tu0, sl1,su1,tl1,tu1] / {100,100,1000,1000,...}
// One wave -> one 16(j) x 64(e) tile; K=8 done as two chained 16x16x4 WMMAs.
__global__ __launch_bounds__(256) void delta_kernel(
    const float* __restrict__ mat,
    const int* __restrict__ traj_len,
    const float* __restrict__ emb_su,
    const float* __restrict__ emb_sl,
    const float* __restrict__ emb_tu,
    const float* __restrict__ emb_tl,
    float* __restrict__ out) {
  const int lane = threadIdx.x & 31;
  const int wave = (blockIdx.x << 3) | (threadIdx.x >> 5);   // 0..32767
  const int b  = wave >> 10;
  const int i  = (wave >> 3) & (kN - 1);
  const int j0 = (wave & 7) << 4;

  const int lo = lane & 15;   // M (row j0+lo) for A, N (col) for B/C/D
  const int hi = lane >> 4;   // selects K pair {2*hi, 2*hi+1}

  const int len = traj_len[b];
  const int jr  = j0 + lo;
  const float fm = ((i < len) && (jr < len)) ? 1.0f : 0.0f;
  const float nm = 1.0f - fm;

  // Per-row interval scalars: this lane's row is jr (duplicated across halves).
  const v2f dsdt = *(const v2f*)(mat + (((size_t)(b * kN + i)) * kN + jr) * 2);
  const float ds = dsdt.x, dt = dsdt.y;

  // k%4 order: 0:vsu 1:vsl 2:vtu 3:vtl. Lane supplies klocal = {2*hi, 2*hi+1}.
  const float c0 = hi ? (1000.0f - dt) : (100.0f - ds);
  const float c1 = hi ? dt : ds;
  // A-matrix f32 16x4 layout: lane L, VGPR v -> (M = L%16, K = 2*(L/16)+v)
  const v2f a0 = { nm * c0, nm * c1 };   // K chunk 0 : mask==0 branch
  const v2f a1 = { fm * c0, fm * c1 };   // K chunk 1 : mask==1 branch

  // B-matrix tables for this lane's k-locals (pairs with vsu/vtu and vsl/vtl).
  const float* tA = hi ? emb_tl : emb_sl;
  const float* tB = hi ? emb_tu : emb_su;
  const float sc  = hi ? 0.001f : 0.01f;   // 1/(TU-TL) , 1/(SU-SL)

  const int row_base = (b * kN + i) * kN + j0;
  // C/D layout: lane L, VGPR v -> (M = v + 8*(L/16), N = L%16)
  float* obase = out + ((size_t)(row_base + 8 * hi)) * kE + lo;

#pragma unroll
  for (int nt = 0; nt < 4; ++nt) {          // four 16-wide N tiles cover E=64
    const int e = nt * 16 + lo;
    const v2f b0 = { tA[e] * sc,      tB[e] * sc      };   // mask row 0
    const v2f b1 = { tA[kE + e] * sc, tB[kE + e] * sc };   // mask row 1
    v8f acc = {0.f, 0.f, 0.f, 0.f, 0.f, 0.f, 0.f, 0.f};
    acc = __builtin_amdgcn_wmma_f32_16x16x4_f32(false, a0, false, b0,
                                                (short)0, acc, false, false);
    acc = __builtin_amdgcn_wmma_f32_16x16x4_f32(false, a1, false, b1,
                                                (short)0, acc, false, false);
    float* op = obase + nt * 16;
#pragma unroll
    for (int v = 0; v < 8; ++v) op[(size_t)v * kE] = acc[v];
  }
}

} // namespace

extern "C" void kernel_launch(void* const* d_in, const int* in_sizes, int n_in,
                              void* d_out, int out_size, void* d_ws, size_t ws_size,
                              hipStream_t stream) {
  const int*   traj     = (const int*)d_in[0];
  const float* mat      = (const float*)d_in[1];
  const int*   traj_len = (const int*)d_in[2];
  const float* emb_t    = (const float*)d_in[3];
  const float* emb_l    = (const float*)d_in[4];
  const float* emb_u    = (const float*)d_in[5];
  const float* emb_su   = (const float*)d_in[6];
  const float* emb_sl   = (const float*)d_in[7];
  const float* emb_tu   = (const float*)d_in[8];
  const float* emb_tl   = (const float*)d_in[9];

  float* joint = (float*)d_out;                       // B*N*E floats
  float* delta = joint + (size_t)kB * kN * kE;        // B*N*N*E floats

  // joint: 262144 elems / 4 per thread / 256 per block = 256 blocks
  joint_kernel<<<(kB * kN * kE) / (4 * 256), 256, 0, stream>>>(
      traj, emb_t, emb_l, emb_u, joint);

  // delta: 32768 waves (one 16x64 tile each), 8 waves per block = 4096 blocks
  delta_kernel<<<(kB * kN * (kN / 16)) / 8, 256, 0, stream>>>(
      mat, traj_len, emb_su, emb_sl, emb_tu, emb_tl, delta);
}